// SelfAttention_61804579389661
// MI455X (gfx1250) — compile-verified
//
#include <hip/hip_runtime.h>

// ---------------------------------------------------------------------------
// Reference output analysis:
//   attention = softmax over a size-1 axis  -> identically 1.0
//   context   = sum_s (attention * hidden)  -> sum_s hidden[b,s,u]
// All the tanh/matmul work is dead code w.r.t. the outputs. Optimal kernel is
// a 256 MB streaming reduction (HBM-bound, ~11us at 23.3 TB/s).
// We perform the reduction with V_WMMA_F32_16X16X4_F32 using an all-ones A
// matrix: D[m,n] = sum_k B[k,n] + C[m,n]  (exact f32, layout-robust in K).
// ---------------------------------------------------------------------------

typedef __attribute__((ext_vector_type(2))) float v2f;
typedef __attribute__((ext_vector_type(8))) float v8f;

#define B_      32
#define S_      2048
#define U_      1024
#define SCHUNKS 8
#define SPER    (S_ / SCHUNKS)   // 256 s-rows per wave-task
#define UTILES  (U_ / 16)        // 64 u-tiles of width 16

// Pass 1: one wave32 per (b, u-tile, s-chunk). Each WMMA sums 4 s-rows over
// 16 u-columns into an f32 accumulator (A = ones). 16384 wave-tasks total.
__global__ __launch_bounds__(256) void attn_reduce_pass1(
    const float* __restrict__ hidden, float* __restrict__ partial) {
  const int wave = blockIdx.x * (blockDim.x >> 5) + (threadIdx.x >> 5);
  const int lane = threadIdx.x & 31;

  const int schunk = wave & (SCHUNKS - 1);        // 3 bits
  const int utile  = (wave >> 3) & (UTILES - 1);  // 6 bits
  const int b      = wave >> 9;                   // 0..31

  const int col  = lane & 15;   // u-column within tile (B/D: N = lane%16)
  const int half = lane >> 4;   // which of the two lane-halves

  const int u0 = utile * 16;
  const int s0 = schunk * SPER;

  // Lane reads rows (s0 + 4*i + half) and (s0 + 4*i + 2 + half), column u0+col.
  // Whatever K-permutation the HW assigns to (VGPR, lane-half), the 4 distinct
  // s-rows per column are all summed because A is all ones.
  const float* p = hidden + ((size_t)b * S_ + (size_t)(s0 + half)) * U_ + u0 + col;

  v2f a;
  a.x = 1.0f;
  a.y = 1.0f;
  v8f acc = {};

#pragma unroll 4
  for (int i = 0; i < SPER / 4; ++i) {
    v2f bb;
    bb.x = p[0];
    bb.y = p[2 * U_];
    p += 4 * U_;
    // D = ones(16x4) * B(4x16) + C : every D row = per-column sum of 4 s-rows
    acc = __builtin_amdgcn_wmma_f32_16x16x4_f32(
        /*neg_a=*/false, a, /*neg_b=*/false, bb,
        /*c_mod=*/(short)0, acc, /*reuse_a=*/false, /*reuse_b=*/false);
  }

  // D layout: VGPR0, lanes 0-15 -> (M=0, N=lane). All M rows identical.
  if (lane < 16) {
    partial[(size_t)(b * SCHUNKS + schunk) * U_ + u0 + lane] = acc[0];
  }
}

// Pass 2: deterministic 8-way combine of s-chunk partials -> context (32x1024).
__global__ __launch_bounds__(256) void attn_reduce_pass2(
    const float* __restrict__ partial, float* __restrict__ context) {
  const int t = blockIdx.x * blockDim.x + threadIdx.x;  // 0..32767
  const int b = t >> 10;
  const int u = t & (U_ - 1);
  float s = 0.0f;
#pragma unroll
  for (int k = 0; k < SCHUNKS; ++k)
    s += partial[(size_t)(b * SCHUNKS + k) * U_ + u];
  context[t] = s;
}

// attention = softmax over a length-1 axis == 1.0 everywhere (32x2048).
__global__ __launch_bounds__(256) void attn_fill_ones(float* __restrict__ att) {
  const int t = blockIdx.x * blockDim.x + threadIdx.x;
  att[t] = 1.0f;
}

extern "C" void kernel_launch(void* const* d_in, const int* in_sizes, int n_in,
                              void* d_out, int out_size, void* d_ws, size_t ws_size,
                              hipStream_t stream) {
  (void)in_sizes; (void)n_in; (void)out_size; (void)ws_size;

  // setup_inputs order: s_prev, hidden_states, Ww, Wb, Uw, Ub, Vw, Vb
  const float* hidden = (const float*)d_in[1];  // (32, 2048, 1024) f32

  float* out     = (float*)d_out;
  float* context = out;            // first 32*1024 floats
  float* att     = out + B_ * U_;  // next 32*2048 floats
  float* partial = (float*)d_ws;   // 32 * 8 * 1024 f32 = 1 MB scratch

  // Pass 1: 16384 wave-tasks -> 2048 blocks x 8 waves
  const int waves = B_ * UTILES * SCHUNKS;
  attn_reduce_pass1<<<waves / 8, 256, 0, stream>>>(hidden, partial);

  // Pass 2: 32*1024 context elements
  attn_reduce_pass2<<<(B_ * U_) / 256, 256, 0, stream>>>(partial, context);

  // Attention ones: 32*2048 elements
  attn_fill_ones<<<(B_ * S_) / 256, 256, 0, stream>>>(att);
}